// IntensityBasedEventLoss_5360119185859
// MI455X (gfx1250) — compile-verified
//
#include <hip/hip_runtime.h>
#include <cstdint>

#define BB 8
#define HH 480
#define WW 640
#define CC 5
#define HWX (HH*WW)                       // 307200
#define NPIX 12
#define TPB 256
#define BLKS_PER_IMG (HWX/(TPB*NPIX))     // 100

typedef uint32_t u32x4 __attribute__((ext_vector_type(4)));
typedef uint32_t u32x8 __attribute__((ext_vector_type(8)));

// ---- CDNA5 Tensor Data Mover: 2D tile (rows x 256 f32, row stride = HWX) -> LDS ----
// D# layout per CDNA5 ISA ch.8: group0 = {count/lds_addr/global_addr/type},
// group1 = {data_size, tensor_dim0/1, tile_dim0/1, tensor_dim0_stride}.
__device__ __forceinline__ void tdm_load_2d(uint32_t lds_byte, const float* gptr,
                                            uint32_t rows) {
  const uint64_t ga = (uint64_t)(uintptr_t)gptr;
  u32x4 g0;
  g0[0] = 1u;                                               // count=1, no gather
  g0[1] = lds_byte;                                         // LDS dest (bytes)
  g0[2] = (uint32_t)ga;                                     // global_addr[31:0]
  g0[3] = (uint32_t)((ga >> 32) & 0x01FFFFFFu) | 0x80000000u; // addr[56:32] | type=2
  u32x8 g1;
  g1[0] = 0x20000u;                                         // data_size=2 (4B), mask=0
  g1[1] = ((uint32_t)HWX & 0xFFFFu) << 16;                  // tensor_dim0[15:0]
  g1[2] = ((uint32_t)HWX >> 16) | (rows << 16);             // tensor_dim0[31:16] | tensor_dim1[15:0]
  g1[3] = ((uint32_t)TPB) << 16;                            // tensor_dim1[31:16]=0 | tile_dim0=256
  g1[4] = rows;                                             // tile_dim1=rows, tile_dim2=0
  g1[5] = (uint32_t)HWX;                                    // tensor_dim0_stride[31:0]
  g1[6] = 0u;                                               // stride0[47:32] | stride1[15:0]
  g1[7] = 0u;
  asm volatile("tensor_load_to_lds %0, %1" :: "s"(g0), "s"(g1) : "memory");
}

// ---- CDNA5 async global->LDS copy (ASYNCcnt path), GVS addressing ----
__device__ __forceinline__ void async_copy_b32(void* lds_ptr, const float* gbase,
                                               uint32_t voff_bytes) {
  uint32_t lds = (uint32_t)(uintptr_t)lds_ptr;   // low 32 bits of generic ptr = LDS offset
  asm volatile("global_load_async_to_lds_b32 %0, %1, %2"
               :: "v"(lds), "v"(voff_bytes), "s"(gbase)
               : "memory");
}

__device__ __forceinline__ float gray_tap(const float* __restrict__ im2b, int xi, int yi) {
  if ((unsigned)xi < (unsigned)WW && (unsigned)yi < (unsigned)HH) {
    int idx = yi * WW + xi;
    return 0.299f * im2b[idx] + 0.587f * im2b[idx + HWX] + 0.114f * im2b[idx + 2 * HWX];
  }
  return 0.0f;
}

__global__ void zero_acc_kernel(float* acc) { acc[threadIdx.x] = 0.0f; }

__global__ __launch_bounds__(TPB) void event_loss_main(
    const float* __restrict__ img1, const float* __restrict__ img2,
    const float* __restrict__ flow, const float* __restrict__ ev,
    const float* __restrict__ vm, float* __restrict__ out, float* __restrict__ acc)
{
  __shared__ float s_ev[2][CC][TPB];   // TDM-filled: rows contiguous, 256 f32 each
  __shared__ float s_im[2][3][TPB];    // TDM-filled
  __shared__ float s_vm[2][TPB];       // per-lane async-filled (wave-local slices)
  __shared__ float s_red[32];

  const int tid  = threadIdx.x;
  const int b    = blockIdx.y;
  const int pix0 = blockIdx.x * (TPB * NPIX);

  const float* im1b = img1 + (size_t)b * 3 * HWX;
  const float* im2b = img2 + (size_t)b * 3 * HWX;
  const float* flb  = flow + (size_t)b * 2 * HWX;
  const float* evb  = ev   + (size_t)b * CC * HWX;
  const float* vmb  = vm   + (size_t)b * HWX;
  float* pred = out + 1 + (size_t)b * CC * HWX;

  if (tid < 32) s_red[tid] = 0.0f;

  float Sp[CC] = {0}, Sg[CC] = {0}, Spg[CC] = {0}, Spp[CC] = {0}, Sgg[CC] = {0};
  float mcnt = 0.0f, l1 = 0.0f;

  // prologue: stage chunk 0 (wave0 issues the two TDM descriptors for the block)
  if (tid < 32) {
    tdm_load_2d((uint32_t)(uintptr_t)&s_ev[0][0][0], evb + pix0, CC);
    tdm_load_2d((uint32_t)(uintptr_t)&s_im[0][0][0], im1b + pix0, 3);
  }
  async_copy_b32(&s_vm[0][tid], vmb, (uint32_t)(pix0 + tid) * 4u);
  __syncthreads();   // also publishes s_red init

  for (int i = 0; i < NPIX; ++i) {
    const int s  = i & 1;
    const int ns = s ^ 1;

    if (i + 1 < NPIX) {  // stage next chunk while doing the scattered warp work
      if (tid < 32) {
        tdm_load_2d((uint32_t)(uintptr_t)&s_ev[ns][0][0], evb + pix0 + (i + 1) * TPB, CC);
        tdm_load_2d((uint32_t)(uintptr_t)&s_im[ns][0][0], im1b + pix0 + (i + 1) * TPB, 3);
      }
      async_copy_b32(&s_vm[ns][tid], vmb, (uint32_t)(pix0 + (i + 1) * TPB + tid) * 4u);
    }

    const int pix = pix0 + i * TPB + tid;
    const float fx = flb[pix];
    const float fy = flb[HWX + pix];
    const int   px = pix % WW;
    const int   py = pix / WW;
    const float x  = (float)px + fx;
    const float y  = (float)py + fy;
    const float x0 = floorf(x), y0 = floorf(y);
    const float wx = x - x0,    wy = y - y0;
    const int x0i = (int)x0,    y0i = (int)y0;

    const float t00 = gray_tap(im2b, x0i,     y0i);
    const float t10 = gray_tap(im2b, x0i + 1, y0i);
    const float t01 = gray_tap(im2b, x0i,     y0i + 1);
    const float t11 = gray_tap(im2b, x0i + 1, y0i + 1);
    const float w2  = t00 * (1.f - wx) * (1.f - wy) + t10 * wx * (1.f - wy)
                    + t01 * (1.f - wx) * wy         + t11 * wx * wy;

    // current chunk's transfers are the oldest in flight (per-wave in-order completion)
    if (i + 1 < NPIX) {
      __builtin_amdgcn_s_wait_tensorcnt(2);
      asm volatile("s_wait_asynccnt 1" ::: "memory");
    } else {
      __builtin_amdgcn_s_wait_tensorcnt(0);
      asm volatile("s_wait_asynccnt 0" ::: "memory");
    }
    __syncthreads();   // publish wave0's TDM-landed data to all waves

    const float g1  = 0.299f * s_im[s][0][tid] + 0.587f * s_im[s][1][tid]
                    + 0.114f * s_im[s][2][tid];
    const float nc  = tanhf((w2 - g1) * 5.0f);
    const float pos = fmaxf(nc, 0.0f);
    const float neg = fmaxf(-nc, 0.0f);
    const float m   = s_vm[s][tid] * (((fx * fx + fy * fy) > 0.01f) ? 1.0f : 0.0f);

    float p[CC];
    p[0] = pos;
    p[1] = neg;
    p[2] = 0.25f * pos + 0.75f * neg;   // w = 1/4
    p[3] = 0.50f * pos + 0.50f * neg;   // w = 2/4
    p[4] = 0.75f * pos + 0.25f * neg;   // w = 3/4

    #pragma unroll
    for (int c = 0; c < CC; ++c) {
      __builtin_nontemporal_store(p[c], &pred[(size_t)c * HWX + pix]);  // write-once stream
      const float pm = p[c] * m;
      const float gm = s_ev[s][c][tid] * m;
      Sp[c]  += pm;       Sg[c]  += gm;
      Spg[c] += pm * gm;  Spp[c] += pm * pm;  Sgg[c] += gm * gm;
      l1 += fabsf(pm - gm);
    }
    mcnt += m;

    if (i + 1 < NPIX) __syncthreads();  // all waves done with buf `s` before it is re-filled
  }

  // wave32 butterfly reduction of the 27 per-lane accumulators
  float v[27];
  #pragma unroll
  for (int c = 0; c < CC; ++c) {
    v[c] = Sp[c]; v[5 + c] = Sg[c]; v[10 + c] = Spg[c];
    v[15 + c] = Spp[c]; v[20 + c] = Sgg[c];
  }
  v[25] = mcnt; v[26] = l1;

  #pragma unroll
  for (int j = 0; j < 27; ++j) {
    #pragma unroll
    for (int off = 16; off >= 1; off >>= 1)
      v[j] += __shfl_xor(v[j], off, 32);
  }
  if ((tid & 31) == 0) {
    #pragma unroll
    for (int j = 0; j < 27; ++j) atomicAdd(&s_red[j], v[j]);   // ds_add_f32
  }
  __syncthreads();
  if (tid < 27) atomicAdd(&acc[b * 32 + tid], s_red[tid]);     // global_atomic_add_f32
}

__global__ void event_loss_final(const float* __restrict__ acc, float* __restrict__ out) {
  __shared__ float sc[64];
  const int t = threadIdx.x;
  float term = 0.0f;
  if (t < BB * CC) {
    const int b = t / CC, c = t % CC;
    const float* a = acc + b * 32;
    const float Sp = a[c], Sg = a[5 + c], Spg = a[10 + c];
    const float Spp = a[15 + c], Sgg = a[20 + c], mc = a[25];
    const float N   = (float)HWX;
    const float num = Spg - Sp * Sg / N;
    const float vp  = fmaxf(Spp - Sp * Sp / N, 0.0f);
    const float vg  = fmaxf(Sgg - Sg * Sg / N, 0.0f);
    const float den = sqrtf(vp) * sqrtf(vg) + 1e-6f;
    term = (mc > 0.0f) ? (1.0f - num / den) : 0.0f;
  }
  sc[t] = term;
  __syncthreads();
  for (int s2 = 32; s2 >= 1; s2 >>= 1) {
    if (t < s2) sc[t] += sc[t + s2];
    __syncthreads();
  }
  if (t == 0) {
    float l1 = 0.0f, ms = 0.0f;
    for (int b = 0; b < BB; ++b) { l1 += acc[b * 32 + 26]; ms += acc[b * 32 + 25]; }
    const float mask_sum = (float)CC * ms + 1e-6f;    // mask broadcast over C channels
    out[0] = l1 / mask_sum + 0.5f * (sc[0] / (float)(BB * CC));
  }
}

extern "C" void kernel_launch(void* const* d_in, const int* in_sizes, int n_in,
                              void* d_out, int out_size, void* d_ws, size_t ws_size,
                              hipStream_t stream) {
  (void)in_sizes; (void)n_in; (void)out_size; (void)ws_size;
  const float* img1 = (const float*)d_in[0];
  const float* img2 = (const float*)d_in[1];
  const float* flow = (const float*)d_in[2];
  const float* ev   = (const float*)d_in[3];
  const float* vmsk = (const float*)d_in[4];
  float* out = (float*)d_out;
  float* acc = (float*)d_ws;                       // 8 images * 32 slots = 256 floats

  zero_acc_kernel<<<1, 256, 0, stream>>>(acc);
  event_loss_main<<<dim3(BLKS_PER_IMG, BB), TPB, 0, stream>>>(
      img1, img2, flow, ev, vmsk, out, acc);
  event_loss_final<<<1, 64, 0, stream>>>(acc, out);
}